// FastRecurrentRunner_8564164788366
// MI455X (gfx1250) — compile-verified
//
#include <hip/hip_runtime.h>
#include <hip/hip_bf16.h>

typedef __attribute__((ext_vector_type(2))) float v2f;
typedef __attribute__((ext_vector_type(8))) float v8f;

// GCC-style 16B int vector matching the async-LDS builtin's parameter type,
// plus address-space-qualified flavors for the global/LDS arguments.
typedef int v4i_vs __attribute__((vector_size(16)));
typedef __attribute__((address_space(1))) v4i_vs gas_v4i;   // global
typedef __attribute__((address_space(3))) v4i_vs las_v4i;   // LDS

constexpr int EMBED  = 2048;
constexpr int HIDDEN = 2048;
constexpr int T      = 32768;

// ---- Phase 2 geometry ----
constexpr int NWG2     = 64;              // persistent workgroups (device barrier members)
constexpr int COLS     = HIDDEN / NWG2;   // 32 output columns per WG
constexpr int THREADS2 = 256;
constexpr int KPER     = THREADS2 / COLS; // 8 k-chunks per column
constexpr int KLEN     = HIDDEN / KPER;   // 256 MACs per thread per step

// ---- CDNA5 async memory->LDS path (ASYNCcnt), with safe fallback ----
#if defined(__has_builtin)
#if __has_builtin(__builtin_amdgcn_global_load_async_to_lds_b128)
#define HAVE_ASYNC_LDS 1
#endif
#endif
#ifndef HAVE_ASYNC_LDS
#define HAVE_ASYNC_LDS 0
#endif

__device__ __forceinline__ void async_copy16(const float* __restrict__ gsrc,
                                             float* __restrict__ ldst) {
#if HAVE_ASYNC_LDS
    // global_load_async_to_lds_b128: per-lane 16B memory -> LDS, no VGPR round-trip
    __builtin_amdgcn_global_load_async_to_lds_b128(
        (gas_v4i*)(gsrc),
        (las_v4i*)(ldst),
        /*offset=*/0, /*cpol=*/0);
#else
    *(float4*)ldst = *(const float4*)gsrc;
#endif
}

__device__ __forceinline__ void wait_async_copies() {
#if HAVE_ASYNC_LDS
#if __has_builtin(__builtin_amdgcn_s_wait_asynccnt)
    __builtin_amdgcn_s_wait_asynccnt(0);
#else
    asm volatile("s_wait_asynccnt 0" ::: "memory");
#endif
#endif
}

// -------------------------------------------------------------------------
// Init: zero H[0,:] (h0 = 0) and the device-barrier counter. Runs every
// launch so graph replay is deterministic.
// -------------------------------------------------------------------------
__global__ void rnn_init(float* __restrict__ H, unsigned int* __restrict__ counter) {
    if (threadIdx.x == 0) *counter = 0u;
    for (int i = threadIdx.x; i < HIDDEN; i += blockDim.x) H[i] = 0.0f;
}

// -------------------------------------------------------------------------
// Phase 1: xproj = X[0:T-1] @ Wx + b, written into H rows 1..T-1.
// Full-precision matrix path: V_WMMA_F32_16X16X4_F32.
// Each wave computes a 16x128 strip; 8 waves/block share one M-tile.
// -------------------------------------------------------------------------
__global__ __launch_bounds__(256) void rnn_xproj_wmma(
    const float* __restrict__ X, const float* __restrict__ Wx,
    const float* __restrict__ bias, float* __restrict__ H) {

    const int lane   = threadIdx.x & 31;
    const int wave   = threadIdx.x >> 5;
    const int sid    = blockIdx.x * 8 + wave;   // 0..32767 strip id
    const int mtile  = sid >> 4;                // 0..2047 (16 rows each)
    const int nbase  = (sid & 15) * 128;        // 16 strips of 128 cols
    const int ml     = lane & 15;
    const int kq2    = (lane >> 4) << 1;        // lanes 0-15: K+{0,1}; 16-31: K+{2,3}

    const float* Arow = X + (size_t)(mtile * 16 + ml) * EMBED;

    v8f acc[8] = {};

    for (int k0 = 0; k0 < EMBED; k0 += 4) {
        // A fragment 16x4 (f32): v2f per lane, 8B-aligned
        v2f a = *(const v2f*)(Arow + k0 + kq2);
        const float* Bp = Wx + (size_t)(k0 + kq2) * HIDDEN + nbase + ml;
        #pragma unroll
        for (int t = 0; t < 8; ++t) {
            // B fragment 4x16: VGPR0 = row K (k0+kq2), VGPR1 = row K+1
            v2f bb;
            bb.x = Bp[t * 16];
            bb.y = Bp[t * 16 + HIDDEN];
            acc[t] = __builtin_amdgcn_wmma_f32_16x16x4_f32(
                false, a, false, bb, (short)0, acc[t], false, false);
        }
    }

    // C layout: VGPR r -> M = r + 8*(lane>>4), N = nbase + (lane&15)
    const int rbase = (lane >> 4) * 8;
    #pragma unroll
    for (int t = 0; t < 8; ++t) {
        const int col = nbase + t * 16 + ml;
        const float bv = bias[col];
        #pragma unroll
        for (int r = 0; r < 8; ++r) {
            const int grow = mtile * 16 + rbase + r;   // source row in X
            if (grow < T - 1)                          // only 32767 valid rows
                H[(size_t)(grow + 1) * HIDDEN + col] = acc[t][r] + bv;
        }
    }
}

// -------------------------------------------------------------------------
// Phase 2: sequential recurrence. 64 persistent WGs, each owns 32 columns
// with its Wh slice resident in 256KB of the 320KB LDS for all 32767 steps.
// Weight preload and per-step h staging use GLOBAL_LOAD_ASYNC_TO_LDS_B128.
// Device-wide split barrier per step via agent-scope atomic counter.
// -------------------------------------------------------------------------
__global__ __launch_bounds__(THREADS2) void rnn_recur(
    const float* __restrict__ Wh, float* __restrict__ H,
    unsigned int* __restrict__ counter) {

    extern __shared__ float smem[];
    float* Wh_s = smem;                          // [HIDDEN][COLS], k-major: k*COLS+jj
    float* h_s  = smem + (size_t)HIDDEN * COLS;  // [HIDDEN]
    float* part = h_s + HIDDEN;                  // [THREADS2]

    const int tid = threadIdx.x;
    const int j0  = blockIdx.x * COLS;

    // One-time async preload of this WG's 256KB weight slice.
    // 16B chunks never cross a 32-float row of the slice (COLS % 4 == 0),
    // and j0 is a multiple of 32, so both sides are 16B aligned.
    for (int v = tid; v < (HIDDEN * COLS) / 4; v += THREADS2) {
        const int idx = v * 4;            // float index into slice
        const int k   = idx / COLS;
        const int jj  = idx % COLS;
        async_copy16(Wh + (size_t)k * HIDDEN + j0 + jj, Wh_s + idx);
    }
    wait_async_copies();
    __syncthreads();

    const int jj = tid & (COLS - 1);
    const int c  = tid >> 5;         // k-chunk 0..7
    const int kb = c * KLEN;

    for (int t = 0; t < T - 1; ++t) {
        // Stage h = H[t,:] into LDS (written device-wide last step):
        // 512 x 16B async chunks across 256 threads.
        const float* hrow = H + (size_t)t * HIDDEN;
        #pragma unroll
        for (int v = 0; v < (HIDDEN / 4) / THREADS2; ++v) {
            const int idx = (v * THREADS2 + tid) * 4;
            async_copy16(hrow + idx, h_s + idx);
        }
        wait_async_copies();
        __syncthreads();

        // Partial dot: column j0+jj over k in [kb, kb+KLEN).
        // h_s[kb+i] is wave-uniform (LDS broadcast); Wh_s stride-1 over lanes.
        float sum = 0.0f;
        const float* wp = Wh_s + (size_t)kb * COLS + jj;
        #pragma unroll 8
        for (int i = 0; i < KLEN; ++i)
            sum = __builtin_fmaf(h_s[kb + i], wp[(size_t)i * COLS], sum);
        part[tid] = sum;
        __syncthreads();

        if (tid < COLS) {
            float tot = 0.0f;
            #pragma unroll
            for (int cc = 0; cc < KPER; ++cc) tot += part[cc * COLS + jj];
            float* hp = H + (size_t)(t + 1) * HIDDEN + j0 + jj;
            const float xp = *hp;                 // xproj written by phase 1
            *hp = tanhf(xp + tot);                // overwrite in place with H[t+1]
        }
        // Warm next step's xproj row while we wait at the barrier.
        if (tid < COLS && (t + 2) < T)
            __builtin_prefetch(&H[(size_t)(t + 2) * HIDDEN + j0 + jj], 0, 0);

        // ---- device-wide barrier ----
        __threadfence();             // release this step's H[t+1] writes
        __syncthreads();
        if (tid == 0) {
            __hip_atomic_fetch_add(counter, 1u, __ATOMIC_ACQ_REL,
                                   __HIP_MEMORY_SCOPE_AGENT);
            const unsigned int target = (unsigned int)(t + 1) * (unsigned int)NWG2;
            while (__hip_atomic_load(counter, __ATOMIC_ACQUIRE,
                                     __HIP_MEMORY_SCOPE_AGENT) < target) {
                __builtin_amdgcn_s_sleep(2);
            }
        }
        __syncthreads();
    }
}

// -------------------------------------------------------------------------
extern "C" void kernel_launch(void* const* d_in, const int* in_sizes, int n_in,
                              void* d_out, int out_size, void* d_ws, size_t ws_size,
                              hipStream_t stream) {
    (void)in_sizes; (void)n_in; (void)out_size; (void)ws_size;

    const float* X    = (const float*)d_in[0];  // [T, EMBED]
    const float* Wx   = (const float*)d_in[1];  // [EMBED, HIDDEN]
    const float* Wh   = (const float*)d_in[2];  // [HIDDEN, HIDDEN]
    const float* bias = (const float*)d_in[3];  // [HIDDEN]
    float* H = (float*)d_out;                   // [T, HIDDEN]
    unsigned int* counter = (unsigned int*)d_ws;

    rnn_init<<<1, 256, 0, stream>>>(H, counter);

    // 2048 M-tiles x 16 N-strips = 32768 wave-strips / 8 waves per block
    rnn_xproj_wmma<<<4096, 256, 0, stream>>>(X, Wx, bias, H);

    const size_t smem_bytes =
        ((size_t)HIDDEN * COLS + HIDDEN + THREADS2) * sizeof(float); // ~265KB < 320KB
    (void)hipFuncSetAttribute(reinterpret_cast<const void*>(rnn_recur),
                              hipFuncAttributeMaxDynamicSharedMemorySize,
                              (int)smem_bytes);
    rnn_recur<<<NWG2, THREADS2, smem_bytes, stream>>>(Wh, H, counter);
}